// AttentionModule_69836168233283
// MI455X (gfx1250) — compile-verified
//
#include <hip/hip_runtime.h>
#include <math.h>

#define B_  16
#define S_  128
#define C_  64
#define H_  8
#define HD_ 64
#define D_  512
#define DH_ 256   // D/2

typedef float v2f __attribute__((ext_vector_type(2)));
typedef float v8f __attribute__((ext_vector_type(8)));

__device__ __forceinline__ float warp_sum(float v) {
    for (int off = 16; off > 0; off >>= 1)
        v += __shfl_down(v, off, 32);
    return v;
}

// ---------------------------------------------------------------------------
// Kernel 0: per-head scalars.  A_h = Wq_h . Wk_h,  Bk_h = bq_h . Wk_h
// (the only parts of the rank-1 Q/K projections that survive the softmax)
// ---------------------------------------------------------------------------
__global__ void head_scalars_kernel(const float* __restrict__ Wq,
                                    const float* __restrict__ bq,
                                    const float* __restrict__ Wk,
                                    float* __restrict__ scal) {
    int h = threadIdx.x >> 5;     // one wave32 per head
    int lane = threadIdx.x & 31;
    if (h < H_) {
        float a = 0.f, bk = 0.f;
        for (int j = lane; j < HD_; j += 32) {
            float wk = Wk[h * HD_ + j];
            a  += Wq[h * HD_ + j] * wk;
            bk += bq[h * HD_ + j] * wk;
        }
        a  = warp_sum(a);
        bk = warp_sum(bk);
        if (lane == 0) { scal[h] = a; scal[H_ + h] = bk; }
    }
}

// ---------------------------------------------------------------------------
// Kernel 1: temporal attention per (b,c).  energy collapses to alpha * x_k
// with alpha = scale*(A_h*x_q + Bk_h); online softmax over k; emits
// xa[b,c,s,h] = sum_k attn*x_k  and its mean over s (for the SE squeeze).
// ---------------------------------------------------------------------------
__global__ void attn_kernel(const float* __restrict__ x,
                            const float* __restrict__ scal,
                            float* __restrict__ xa,
                            float* __restrict__ xam) {
    __shared__ float xcol[S_];
    __shared__ float red[H_ * S_];
    int r = blockIdx.x;            // r = b*C + c
    int b = r / C_, c = r % C_;
    int t = threadIdx.x;

    for (int s = t; s < S_; s += 256)
        xcol[s] = x[(b * S_ + s) * C_ + c];
    __syncthreads();

    const float scale = 0.125f;    // 1/sqrt(HD)
    for (int p = 0; p < (H_ * S_) / 256; ++p) {
        int idx = p * 256 + t;
        int h = idx >> 7;          // S_ == 128
        int q = idx & (S_ - 1);
        float alpha = scale * (scal[h] * xcol[q] + scal[H_ + h]);
        float m = -INFINITY, Z = 0.f, W = 0.f;
        for (int k = 0; k < S_; ++k) {
            float xk = xcol[k];
            float e  = alpha * xk;
            float mn = fmaxf(m, e);
            float corr = __expf(m - mn);
            float pe   = __expf(e - mn);
            Z = Z * corr + pe;
            W = W * corr + pe * xk;
            m = mn;
        }
        float val = W / Z;
        xa[(r * S_ + q) * H_ + h] = val;
        red[idx] = val;
    }
    __syncthreads();
    if (t < H_) {
        float s = 0.f;
        for (int q = 0; q < S_; ++q) s += red[t * S_ + q];
        xam[r * H_ + t] = s * (1.0f / (float)S_);
    }
}

// ---------------------------------------------------------------------------
// Kernel 2: sq[r][d] = Wv[d]*xam[r][h(d)] + bv[d]   (SE squeeze input, 1024x512)
// ---------------------------------------------------------------------------
__global__ void sq_kernel(const float* __restrict__ xam,
                          const float* __restrict__ Wv,
                          const float* __restrict__ bv,
                          float* __restrict__ sq) {
    int idx = blockIdx.x * blockDim.x + threadIdx.x;
    if (idx >= (B_ * C_) * D_) return;
    int r = idx / D_, d = idx % D_;
    sq[idx] = Wv[d] * xam[r * H_ + (d / HD_)] + bv[d];
}

// ---------------------------------------------------------------------------
// WMMA GEMM (fp32): C[M,N] = A[M,K] @ B[K,N] + bias (optionally sigmoid).
// Block = 8 waves = 128x16 output panel.  The Kx16 B panel is staged in LDS
// once per block via CDNA5 async global->LDS copies (ASYNCcnt), then each
// wave runs V_WMMA_F32_16X16X4_F32 along K with:
//   A frag (16x4): lanes 0-15 -> K={k,k+1}, lanes 16-31 -> K={k+2,k+3}
//   B frag (4x16): VGPR0 rows {k,k+2}, VGPR1 rows {k+1,k+3}, N = lane%16
//   C/D (16x16):   VGPR i -> row i (lanes 0-15) / row i+8 (lanes 16-31)
// Bp[k*16 + l15] puts the 16 lanes in 16 distinct banks; the two half-wave
// K-offsets differ by 32 banks -> conflict-free ds reads.
// ---------------------------------------------------------------------------
__global__ __launch_bounds__(256) void gemm_wmma_kernel(
        const float* __restrict__ A, const float* __restrict__ B,
        const float* __restrict__ bias, float* __restrict__ C,
        int M, int N, int K, int do_sigmoid) {
    extern __shared__ float Bp[];          // K x 16 floats

    int t = threadIdx.x;
    int wave = t >> 5, lane = t & 31;
    int nt = N >> 4;
    int bm = blockIdx.x / nt;              // 128-row panel index
    int n0 = (blockIdx.x % nt) << 4;       // 16-col tile

    // --- stage B panel: global -> LDS, 16 bytes per lane per async op ---
    int quads = (K * 16) >> 2;             // # of 4-float chunks
    for (int e = t; e < quads; e += 256) {
        int k  = e >> 2;
        int cg = (e & 3) << 2;
        unsigned lds_addr = (unsigned)(uintptr_t)(&Bp[k * 16 + cg]);
        const float* gp = B + (size_t)k * N + n0 + cg;
        asm volatile("global_load_async_to_lds_b128 %0, %1, off"
                     :: "v"(lds_addr), "v"(gp)
                     : "memory");
    }
    asm volatile("s_wait_asynccnt 0x0" ::: "memory");  // my transfers done
    __syncthreads();                                    // everyone's visible

    // --- WMMA main loop ---
    int tile_m = (bm << 7) + (wave << 4);
    int half = lane >> 4;                  // 0: lanes 0-15, 1: lanes 16-31
    int l15  = lane & 15;

    v8f acc = {};
    const float* arow = A + (size_t)(tile_m + l15) * K;
    #pragma unroll 4
    for (int k = 0; k < K; k += 4) {
        int ka = k + half * 2;
        v2f a, b;
        a.x = arow[ka];
        a.y = arow[ka + 1];
        b.x = Bp[ka * 16 + l15];
        b.y = Bp[(ka + 1) * 16 + l15];
        acc = __builtin_amdgcn_wmma_f32_16x16x4_f32(
            /*neg_a=*/false, a, /*neg_b=*/false, b,
            /*c_mod=*/(short)0, acc, /*reuse_a=*/false, /*reuse_b=*/false);
    }

    int col = n0 + l15;
    float bsv = bias[col];
    #pragma unroll
    for (int i = 0; i < 8; ++i) {
        int row = tile_m + i + half * 8;
        float v = acc[i] + bsv;
        if (do_sigmoid) v = 1.0f / (1.0f + __expf(-v));
        C[(size_t)row * N + col] = v;
    }
}

// ---------------------------------------------------------------------------
// Kernel 5: final projection.
// final[b,s,c] = sum_h xa[b,c,s,h]*g[b,c,h] + t[b,c] + bf
//   g[b,c,h] = sum_hd Wv[d]*exc[b,c,d]*Wf[d],  t[b,c] = sum_d bv[d]*exc*Wf
// ---------------------------------------------------------------------------
__global__ void final_kernel(const float* __restrict__ xa,
                             const float* __restrict__ exc,
                             const float* __restrict__ Wv,
                             const float* __restrict__ bv,
                             const float* __restrict__ Wf,
                             const float* __restrict__ bf,
                             float* __restrict__ out) {
    __shared__ float gprod[D_];
    __shared__ float tprod[D_];
    __shared__ float gs[H_];
    __shared__ float th[H_];
    __shared__ float tsum;
    int r = blockIdx.x;
    int b = r / C_, c = r % C_;
    int t = threadIdx.x;

    for (int d = t; d < D_; d += 256) {
        float e  = exc[(size_t)r * D_ + d];
        float wf = Wf[d];
        gprod[d] = Wv[d] * e * wf;
        tprod[d] = bv[d] * e * wf;
    }
    __syncthreads();
    if (t < H_) {
        float g = 0.f, tv = 0.f;
        for (int j = 0; j < HD_; ++j) {
            g  += gprod[t * HD_ + j];
            tv += tprod[t * HD_ + j];
        }
        gs[t] = g; th[t] = tv;
    }
    __syncthreads();
    if (t == 0) {
        float s = 0.f;
        for (int h = 0; h < H_; ++h) s += th[h];
        tsum = s;
    }
    __syncthreads();
    for (int s = t; s < S_; s += 256) {
        float acc = tsum + bf[0];
        const float* xap = &xa[(size_t)(r * S_ + s) * H_];
        #pragma unroll
        for (int h = 0; h < H_; ++h) acc += xap[h] * gs[h];
        out[(b * S_ + s) * C_ + c] = acc;
    }
}

// ---------------------------------------------------------------------------
extern "C" void kernel_launch(void* const* d_in, const int* in_sizes, int n_in,
                              void* d_out, int out_size, void* d_ws, size_t ws_size,
                              hipStream_t stream) {
    const float* x  = (const float*)d_in[0];
    const float* Wq = (const float*)d_in[1];
    const float* bq = (const float*)d_in[2];
    const float* Wk = (const float*)d_in[3];
    const float* bk = (const float*)d_in[4];  (void)bk; // cancels in softmax
    const float* Wv = (const float*)d_in[5];
    const float* bv = (const float*)d_in[6];
    const float* Ws = (const float*)d_in[7];
    const float* bs = (const float*)d_in[8];
    const float* We = (const float*)d_in[9];
    const float* be = (const float*)d_in[10];
    const float* Wf = (const float*)d_in[11];
    const float* bf = (const float*)d_in[12];
    float* out = (float*)d_out;

    // workspace carve-up (floats)
    float* p    = (float*)d_ws;
    float* scal = p; p += 16;
    float* xa   = p; p += (size_t)B_ * C_ * S_ * H_;   // 1,048,576
    float* xam  = p; p += (size_t)B_ * C_ * H_;        // 8,192
    float* sq   = p; p += (size_t)B_ * C_ * D_;        // 524,288
    float* hid  = p; p += (size_t)B_ * C_ * DH_;       // 262,144
    float* exc  = p; p += (size_t)B_ * C_ * D_;        // 524,288

    head_scalars_kernel<<<1, 256, 0, stream>>>(Wq, bq, Wk, scal);
    attn_kernel<<<B_ * C_, 256, 0, stream>>>(x, scal, xa, xam);

    int sqN = B_ * C_ * D_;
    sq_kernel<<<(sqN + 255) / 256, 256, 0, stream>>>(xam, Wv, bv, sq);

    // GEMM1: (1024 x 512) @ Ws(512 x 256) + bs -> hid
    {
        int blocks = (B_ * C_ / 128) * (DH_ / 16);     // 8*16 = 128
        size_t shmem = (size_t)D_ * 16 * sizeof(float); // 32 KB B panel
        gemm_wmma_kernel<<<blocks, 256, shmem, stream>>>(sq, Ws, bs, hid,
                                                         B_ * C_, DH_, D_, 0);
    }
    // GEMM2: (1024 x 256) @ We(256 x 512) + be, sigmoid -> exc
    {
        int blocks = (B_ * C_ / 128) * (D_ / 16);      // 8*32 = 256
        size_t shmem = (size_t)DH_ * 16 * sizeof(float); // 16 KB B panel
        gemm_wmma_kernel<<<blocks, 256, shmem, stream>>>(hid, We, be, exc,
                                                         B_ * C_, D_, DH_, 1);
    }
    final_kernel<<<B_ * C_, 256, 0, stream>>>(xa, exc, Wv, bv, Wf, bf, out);
}